// GatedBlock_85555748536464
// MI455X (gfx1250) — compile-verified
//
#include <hip/hip_runtime.h>
#include <math.h>

#define NN 50000
#define EE 800000
#define DD 128
#define TM 16   // rows per workgroup tile (50000 = 16 * 3125, exact)

typedef __attribute__((ext_vector_type(2))) float v2f;
typedef __attribute__((ext_vector_type(8))) float v8f;
typedef __attribute__((ext_vector_type(4))) unsigned int v4u;
typedef __attribute__((ext_vector_type(8))) int v8i;
typedef __attribute__((ext_vector_type(4))) int v4i;

#if defined(__has_builtin)
#  if __has_builtin(__builtin_amdgcn_tensor_load_to_lds)
#    define HAVE_TDM 1
#  endif
#endif
#ifndef HAVE_TDM
#  define HAVE_TDM 0
#endif

// ---------------------------------------------------------------------------
// Kernel 1: zero workspace (summed[N*D] + counts[N])
// ---------------------------------------------------------------------------
__global__ __launch_bounds__(256) void zero_ws_kernel(float* __restrict__ ws, int total) {
    int i = blockIdx.x * 256 + threadIdx.x;
    if (i < total) ws[i] = 0.0f;
}

// ---------------------------------------------------------------------------
// Kernel 2: scatter-mean accumulation.
// One thread per (edge, feature). 256 threads = 2 edges/block (coalesced 128B
// per wave for both the x[src] gather and the summed[dst] atomic — both
// L2-resident on MI455X (x = 25.6MB, summed = 25.6MB, L2 = 192MB)).
// ---------------------------------------------------------------------------
__global__ __launch_bounds__(256) void scatter_mean_kernel(
    const float* __restrict__ x,
    const long long* __restrict__ edge_index,   // [2, E] int64
    float* __restrict__ summed,                 // [N, D]
    float* __restrict__ counts)                 // [N]
{
    const int tid = threadIdx.x;
    const int e = blockIdx.x * 2 + (tid >> 7);   // 2 edges per block
    const int d = tid & 127;
    const int src = (int)edge_index[e];
    const int dst = (int)edge_index[EE + e];
    atomicAdd(&summed[dst * DD + d], x[src * DD + d]);
    if (d == 0) atomicAdd(&counts[dst], 1.0f);
}

// ---------------------------------------------------------------------------
// TDM: DMA a TM x DD f32 row-major tile (stride DD) from global into LDS.
// D# layout per CDNA5 ISA §8:
//   group0: [1:0]=count=1, [63:32]=lds_addr, [120:64]=global_addr, [127:126]=type=2
//   group1: [17:16]=data_size(2=4B), [79:48]=tensor_dim0, [111:80]=tensor_dim1,
//           [127:112]=tile_dim0, [143:128]=tile_dim1, [207:160]=tensor_dim0_stride
//   groups 2/3: zero (2D tensor)
// Issued by one wave; completion via s_wait_tensorcnt; published via barrier.
// ---------------------------------------------------------------------------
#if HAVE_TDM
__device__ __forceinline__ void tdm_load_tile(const float* gptr, unsigned lds_off) {
    const unsigned long long ga = (unsigned long long)(uintptr_t)gptr;
    v4u g0;
    g0.x = 0x1u;                                           // count=1, user mode
    g0.y = lds_off;                                        // lds_addr (bytes)
    g0.z = (unsigned)(ga & 0xffffffffu);                   // global_addr[31:0]
    g0.w = (unsigned)((ga >> 32) & 0x01ffffffu) | (2u << 30); // addr[56:32] | type=2

    v8i g1;
    g1[0] = (int)(2u << 16);                               // wg_mask=0, data_size=4B
    g1[1] = (int)((DD & 0xffffu) << 16);                   // tensor_dim0[15:0] -> bits[63:48]
    g1[2] = (int)(((unsigned)DD >> 16) | ((TM & 0xffffu) << 16)); // tdim0 hi | tensor_dim1 lo
    g1[3] = (int)(((unsigned)TM >> 16) | ((unsigned)DD << 16));   // tdim1 hi | tile_dim0=DD
    g1[4] = (int)(TM);                                     // tile_dim1=TM, tile_dim2=0
    g1[5] = (int)(DD);                                     // tensor_dim0_stride[31:0]
    g1[6] = 0;                                             // stride hi | dim1_stride lo
    g1[7] = 0;

    const v4i z4 = {0, 0, 0, 0};
#if defined(__clang_major__) && __clang_major__ >= 23
    const v8i z8 = {0, 0, 0, 0, 0, 0, 0, 0};
    __builtin_amdgcn_tensor_load_to_lds(g0, g1, z4, z4, z8, 0);
#else
    __builtin_amdgcn_tensor_load_to_lds(g0, g1, z4, z4, 0);
#endif
}
#endif

// ---------------------------------------------------------------------------
// Kernel 3: fused  h = aggr@W_l + b_l + x@W_r ; BN ; GELU ;
//                  gate = sigmoid([x,h]@W_g + b_g) ; out = gate*x+(1-gate)*h
// 256 threads = 8 waves. Wave w computes output columns [16w, 16w+16).
// All GEMMs via V_WMMA_F32_16X16X4_F32 (f32 in, f32 acc).
// ---------------------------------------------------------------------------
__device__ __forceinline__ v8f wmma_f32(v2f a, v2f b, v8f c) {
    return __builtin_amdgcn_wmma_f32_16x16x4_f32(
        /*neg_a=*/false, a, /*neg_b=*/false, b,
        /*c_mod=*/(short)0, c, /*reuse_a=*/false, /*reuse_b=*/false);
}

__global__ __launch_bounds__(256) void fused_sage_kernel(
    const float* __restrict__ x,
    const float* __restrict__ summed,
    const float* __restrict__ counts,
    const float* __restrict__ W_l, const float* __restrict__ b_l,
    const float* __restrict__ W_r,
    const float* __restrict__ bn_gamma, const float* __restrict__ bn_beta,
    const float* __restrict__ bn_mean,  const float* __restrict__ bn_var,
    const float* __restrict__ W_g, const float* __restrict__ b_g,
    float* __restrict__ out)
{
    __shared__ float xs[TM * DD];   // x tile        (8 KB)
    __shared__ float as[TM * DD];   // aggr tile     (8 KB)
    __shared__ float hs[TM * DD];   // h tile        (8 KB)

    const int row0 = blockIdx.x * TM;
    const int tid  = threadIdx.x;

    // L2 warm-up hints for the (shared, hot) weight matrices.
    if (tid == 0) {
        __builtin_prefetch(W_l, 0, 2);
        __builtin_prefetch(W_r, 0, 2);
        __builtin_prefetch(W_g, 0, 2);
        __builtin_prefetch(W_g + DD * DD, 0, 2);
    }

#if HAVE_TDM
    // --- Async tile staging via the Tensor Data Mover (wave 0 issues DMA) ---
    if (tid < 32) {
        tdm_load_tile(x + (size_t)row0 * DD,      (unsigned)(uintptr_t)&xs[0]);
        tdm_load_tile(summed + (size_t)row0 * DD, (unsigned)(uintptr_t)&as[0]);
        __builtin_amdgcn_s_wait_tensorcnt(0);
    }
    __syncthreads();
    // In-place count divide: aggr = summed / max(count, 1)
    for (int i = tid; i < TM * DD; i += 256) {
        const int m = i >> 7;
        as[i] *= 1.0f / fmaxf(counts[row0 + m], 1.0f);
    }
    __syncthreads();
#else
    // --- Fallback: manual staging through VGPRs ---
    for (int i = tid; i < TM * DD; i += 256) {
        const int m = i >> 7;
        const float inv = 1.0f / fmaxf(counts[row0 + m], 1.0f);
        xs[i] = x[row0 * DD + i];
        as[i] = summed[row0 * DD + i] * inv;
    }
    __syncthreads();
#endif

    const int wave  = tid >> 5;               // 0..7
    const int lane  = tid & 31;
    const int hi    = lane >> 4;              // 0 (lanes 0-15) / 1 (lanes 16-31)
    const int ncol  = wave * 16 + (lane & 15);// this lane's output column
    const int arow  = lane & 15;              // A-fragment row (M)

    // ---- GEMM 1: acc = aggr@W_l + x@W_r  (K = 128, 32 WMMA steps x2) ----
    v8f acc = {};
    #pragma unroll 4
    for (int k0 = 0; k0 < DD; k0 += 4) {
        const int ka = k0 + 2 * hi;           // A/B layout: lanes16-31 hold K=2,3
        v2f a, b;
        a.x = as[arow * DD + ka];     a.y = as[arow * DD + ka + 1];
        b.x = W_l[ka * DD + ncol];    b.y = W_l[(ka + 1) * DD + ncol];
        acc = wmma_f32(a, b, acc);
        a.x = xs[arow * DD + ka];     a.y = xs[arow * DD + ka + 1];
        b.x = W_r[ka * DD + ncol];    b.y = W_r[(ka + 1) * DD + ncol];
        acc = wmma_f32(a, b, acc);
    }

    // ---- epilogue 1: +b_l, BatchNorm(eval), exact GELU; h -> LDS ----
    const float scale = bn_gamma[ncol] * rsqrtf(bn_var[ncol] + 1e-5f);
    const float shift = bn_beta[ncol] - bn_mean[ncol] * scale;
    const float bias  = b_l[ncol];
    #pragma unroll
    for (int i = 0; i < 8; ++i) {
        float h = (acc[i] + bias) * scale + shift;
        h = 0.5f * h * (1.0f + erff(h * 0.70710678118654752f));
        const int m = i + 8 * hi;             // C/D layout: VGPR i -> M=i (+8 for hi lanes)
        hs[m * DD + ncol] = h;
    }
    __syncthreads();

    // ---- GEMM 2: g = x@W_g[0:128] + h@W_g[128:256] ----
    v8f g = {};
    #pragma unroll 4
    for (int k0 = 0; k0 < DD; k0 += 4) {
        const int ka = k0 + 2 * hi;
        v2f a, b;
        a.x = xs[arow * DD + ka];           a.y = xs[arow * DD + ka + 1];
        b.x = W_g[ka * DD + ncol];          b.y = W_g[(ka + 1) * DD + ncol];
        g = wmma_f32(a, b, g);
        a.x = hs[arow * DD + ka];           a.y = hs[arow * DD + ka + 1];
        b.x = W_g[(DD + ka) * DD + ncol];   b.y = W_g[(DD + ka + 1) * DD + ncol];
        g = wmma_f32(a, b, g);
    }

    // ---- epilogue 2: sigmoid gate + blend, store ----
    const float bg = b_g[ncol];
    #pragma unroll
    for (int i = 0; i < 8; ++i) {
        const int m = i + 8 * hi;
        const float z    = g[i] + bg;
        const float gate = 1.0f / (1.0f + expf(-z));
        const float xv   = xs[m * DD + ncol];
        const float hv   = hs[m * DD + ncol];
        out[(row0 + m) * DD + ncol] = gate * xv + (1.0f - gate) * hv;
    }
}

// ---------------------------------------------------------------------------
extern "C" void kernel_launch(void* const* d_in, const int* in_sizes, int n_in,
                              void* d_out, int out_size, void* d_ws, size_t ws_size,
                              hipStream_t stream) {
    const float*     x        = (const float*)d_in[0];
    const long long* edge_idx = (const long long*)d_in[1];
    const float*     W_l      = (const float*)d_in[2];
    const float*     b_l      = (const float*)d_in[3];
    const float*     W_r      = (const float*)d_in[4];
    const float*     bn_gamma = (const float*)d_in[5];
    const float*     bn_beta  = (const float*)d_in[6];
    const float*     bn_mean  = (const float*)d_in[7];
    const float*     bn_var   = (const float*)d_in[8];
    const float*     W_g      = (const float*)d_in[9];
    const float*     b_g      = (const float*)d_in[10];
    float*           out      = (float*)d_out;

    float* summed = (float*)d_ws;             // N*D floats
    float* counts = summed + (size_t)NN * DD; // N floats

    const int zero_total = NN * DD + NN;
    zero_ws_kernel<<<(zero_total + 255) / 256, 256, 0, stream>>>(summed, zero_total);

    scatter_mean_kernel<<<EE / 2, 256, 0, stream>>>(x, edge_idx, summed, counts);

    fused_sage_kernel<<<NN / TM, 256, 0, stream>>>(
        x, summed, counts, W_l, b_l, W_r,
        bn_gamma, bn_beta, bn_mean, bn_var, W_g, b_g, out);
}